// RandomDE_29901562315443
// MI455X (gfx1250) — compile-verified
//
#include <hip/hip_runtime.h>
#include <stdint.h>

#define TPB 256
#define ROWS 16
#define CPT 4                       // columns per thread
#define COLS_PER_BLOCK (TPB * CPT)  // 1024

typedef __attribute__((ext_vector_type(4))) float v4f;
typedef __attribute__((ext_vector_type(4))) int   v4i;

// ---- CDNA5 async global->LDS copy helpers (ASYNCcnt path) ------------------
__device__ __forceinline__ void async_g2l_b128(uint32_t lds_off, const void* gaddr) {
    asm volatile("global_load_async_to_lds_b128 %0, %1, off"
                 :: "v"(lds_off), "v"(gaddr) : "memory");
}
__device__ __forceinline__ void async_g2l_b32(uint32_t lds_off, const void* gaddr) {
    asm volatile("global_load_async_to_lds_b32 %0, %1, off"
                 :: "v"(lds_off), "v"(gaddr) : "memory");
}
__device__ __forceinline__ void wait_async0() {
#if __has_builtin(__builtin_amdgcn_s_wait_asynccnt)
    __builtin_amdgcn_s_wait_asynccnt(0);
#else
    asm volatile("s_wait_asynccnt 0" ::: "memory");
#endif
}

// ORDER = 2 (pair terms) or 3 (triple terms).
// Block: TPB threads, handles ROWS batch rows x COLS_PER_BLOCK output columns.
template <int ORDER>
__global__ __launch_bounds__(TPB) void poly_gather_kernel(
    const float* __restrict__ x,    // B x D, row-major
    const int*   __restrict__ idx,  // O x ORDER, row-major
    float* __restrict__ out,        // B x OUT_W
    int D, int OUT_W, int colOfs)
{
    extern __shared__ float smem[];       // ROWS * D floats
    const int tid  = threadIdx.x;
    const int row0 = blockIdx.y * ROWS;
    const int col0 = blockIdx.x * COLS_PER_BLOCK + tid * CPT;

    // ---- stage ROWS contiguous rows of x into LDS via async b128 copies ----
    const float* gsrc = x + (size_t)row0 * D;
    const uint32_t ldsBase = (uint32_t)(uintptr_t)smem;  // low 32 bits = LDS offset
    const int totalFloats = ROWS * D;
    const int totalVec = totalFloats >> 2;               // 16B chunks
    for (int v = tid; v < totalVec; v += TPB) {
        async_g2l_b128(ldsBase + (uint32_t)(v * 16), (const void*)(gsrc + v * 4));
    }
    // tail (never runs: ROWS * D is a multiple of 4)
    for (int f = (totalVec << 2) + tid; f < totalFloats; f += TPB) {
        async_g2l_b32(ldsBase + (uint32_t)(f * 4), (const void*)(gsrc + f));
    }

    // ---- load this thread's column indices (contiguous -> wide loads) ------
    int cidx[CPT * ORDER];
    {
        const v4i* ip = reinterpret_cast<const v4i*>(idx + (size_t)col0 * ORDER);
        v4i* cp = reinterpret_cast<v4i*>(cidx);
#pragma unroll
        for (int k = 0; k < (CPT * ORDER) / 4; ++k) cp[k] = ip[k];
    }

    wait_async0();        // our wave's async writes to LDS are done
    __syncthreads();      // everyone's writes visible

    // ---- gather from LDS, multiply, stream out with NT b128 stores ---------
    float* obase = out + colOfs + (size_t)blockIdx.x * COLS_PER_BLOCK + tid * CPT;
#pragma unroll 2
    for (int r = 0; r < ROWS; ++r) {
        const float* xr = smem + r * D;
        float p[CPT];
#pragma unroll
        for (int c = 0; c < CPT; ++c) {
            float v = xr[cidx[c * ORDER + 0]] * xr[cidx[c * ORDER + 1]];
            if (ORDER == 3) v *= xr[cidx[c * ORDER + 2]];
            p[c] = v;
        }
        v4f vec = { p[0], p[1], p[2], p[3] };
        float* dst = obase + (size_t)(row0 + r) * OUT_W;
        __builtin_nontemporal_store(vec, reinterpret_cast<v4f*>(dst));
    }
}

extern "C" void kernel_launch(void* const* d_in, const int* in_sizes, int n_in,
                              void* d_out, int out_size, void* d_ws, size_t ws_size,
                              hipStream_t stream) {
    const float* x    = (const float*)d_in[0];
    const int*   idx1 = (const int*)d_in[1];
    const int*   idx2 = (const int*)d_in[2];
    float* out = (float*)d_out;

    const int O1    = in_sizes[1] / 2;        // 8192
    const int O2    = in_sizes[2] / 3;        // 8192
    const int OUT_W = O1 + O2;                // 16384
    const int B     = out_size / OUT_W;       // 4096
    const int D     = in_sizes[0] / B;        // 784

    const size_t ldsBytes = (size_t)ROWS * D * sizeof(float);  // ~49 KB
    dim3 block(TPB);

    dim3 grid1(O1 / COLS_PER_BLOCK, B / ROWS);   // (8, 256)
    poly_gather_kernel<2><<<grid1, block, ldsBytes, stream>>>(x, idx1, out, D, OUT_W, 0);

    dim3 grid2(O2 / COLS_PER_BLOCK, B / ROWS);   // (8, 256)
    poly_gather_kernel<3><<<grid2, block, ldsBytes, stream>>>(x, idx2, out, D, OUT_W, O1);
}